// TTA_927712936423
// MI455X (gfx1250) — compile-verified
//
#include <hip/hip_runtime.h>

typedef __attribute__((ext_vector_type(2))) float v2f;
typedef __attribute__((ext_vector_type(8))) float v8f;

#define DEV_INLINE __device__ __forceinline__

// ---------------------------------------------------------------------------
// Constant prior index table (204 entries, order F1,F2,F3,F4,F5,F0)
// ---------------------------------------------------------------------------
__constant__ int PRIOR_IDX[204] = {
    // F1 (36)
    697, 698, 699, 700, 712, 713, 714, 715, 737, 738, 739, 740, 741, 743, 744,
    745, 746, 748, 749, 750, 753, 754, 755, 756, 757, 758, 759, 760, 761, 762,
    763, 764, 765, 766, 767, 768,
    // F2 (46)
    46, 47, 48, 49, 164, 165, 166, 167, 194, 195, 223, 237, 238, 280, 281,
    298, 301, 317, 320, 323, 324, 325, 326, 327, 328, 329, 330, 331, 332, 333,
    340, 341, 342, 343, 344, 345, 346, 347, 348, 349, 350, 351, 352, 353, 354,
    355,
    // F3 (40)
    356, 357, 358, 359, 375, 376, 386, 387, 396, 397, 402, 403, 413, 429, 433,
    434, 435, 436, 437, 438, 439, 440, 441, 442, 443, 444, 452, 453, 454, 455,
    456, 459, 460, 461, 462, 463, 464, 465, 466, 467,
    // F4 (37)
    468, 469, 470, 471, 484, 485, 486, 496, 497, 506, 507, 513, 514, 524, 545,
    546, 547, 548, 549, 550, 551, 552, 553, 555, 563, 564, 565, 566, 567, 570,
    572, 573, 574, 575, 576, 577, 578,
    // F5 (37)
    580, 581, 582, 583, 600, 601, 602, 614, 615, 624, 625, 630, 631, 641, 663,
    664, 665, 666, 667, 668, 670, 672, 680, 681, 682, 683, 684, 686, 687, 688,
    689, 690, 691, 692, 693, 694, 695,
    // F0 (8)
    73, 96, 98, 99, 772, 774, 775, 777};

DEV_INLINE v8f wmma_16x16x4_f32(v2f a, v2f b, v8f c) {
  // 8-arg form: (neg_a, A, neg_b, B, c_mod, C, reuse_a, reuse_b)
  return __builtin_amdgcn_wmma_f32_16x16x4_f32(false, a, false, b, (short)0, c,
                                               false, false);
}

// ---------------------------------------------------------------------------
// init: zero vn + accumulators, set colmin to +inf
// ---------------------------------------------------------------------------
__global__ void init_kernel(float* __restrict__ vn, unsigned* __restrict__ colmin,
                            float* __restrict__ acc, int nVn, int nCol) {
  int i = blockIdx.x * blockDim.x + threadIdx.x;
  if (i < nVn) vn[i] = 0.f;
  if (i < nCol) colmin[i] = 0x7F800000u;  // +inf
  if (i < 16) acc[i] = 0.f;
}

// ---------------------------------------------------------------------------
// gather prior points: prior[b, j] = recon[b, PRIOR_IDX[j]]
// ---------------------------------------------------------------------------
__global__ void gather_prior(const float* __restrict__ recon,
                             float* __restrict__ prior, int B, int V, int NP) {
  int i = blockIdx.x * blockDim.x + threadIdx.x;
  if (i >= B * NP) return;
  int b = i / NP, j = i % NP;
  int vi = PRIOR_IDX[j];
  const float* s = recon + ((size_t)b * V + vi) * 3;
  float* d = prior + (size_t)i * 3;
  d[0] = s[0]; d[1] = s[1]; d[2] = s[2];
}

// ---------------------------------------------------------------------------
// Scan all 16-column target tiles against a fixed 16x4 A tile.
// Branch-free main loop (full tiles) + one clamped-load tail tile.
// ---------------------------------------------------------------------------
template <bool WITH_IDX>
DEV_INLINE void knn_scan(v2f A, const float a2v[8],
                         const float* __restrict__ tb, int Mt, int l, int half,
                         float dmin[8], int jmin[8]) {
#pragma unroll
  for (int r = 0; r < 8; ++r) {
    dmin[r] = __builtin_inff();
    if (WITH_IDX) jmin[r] = 0;
  }
  const int full = Mt >> 4;
  for (int ct = 0; ct < full; ++ct) {
    const int col = (ct << 4) + l;
    const float* q = tb + (size_t)col * 3;
    float tx = q[0], ty = q[1], tz = q[2];
    float b2 = tx * tx + ty * ty + tz * tz;
    v2f Bv;
    Bv.x = half ? tz : tx;
    Bv.y = half ? 0.f : ty;
    v8f c = {0.f, 0.f, 0.f, 0.f, 0.f, 0.f, 0.f, 0.f};
    v8f ab = wmma_16x16x4_f32(A, Bv, c);
#pragma unroll
    for (int r = 0; r < 8; ++r) {
      float d2 = fmaxf(a2v[r] + b2 - 2.f * ab[r], 0.f);
      if (WITH_IDX) {
        if (d2 < dmin[r]) { dmin[r] = d2; jmin[r] = col; }
      } else {
        dmin[r] = fminf(dmin[r], d2);
      }
    }
  }
  if (Mt & 15) {  // single tail tile, clamped load (no EXEC divergence)
    const int col = (full << 4) + l;
    const bool valid = col < Mt;
    const float* q = tb + (size_t)(valid ? col : (Mt - 1)) * 3;
    float tx = q[0], ty = q[1], tz = q[2];
    float b2 = valid ? (tx * tx + ty * ty + tz * tz) : __builtin_inff();
    v2f Bv;
    Bv.x = half ? tz : tx;
    Bv.y = half ? 0.f : ty;
    v8f c = {0.f, 0.f, 0.f, 0.f, 0.f, 0.f, 0.f, 0.f};
    v8f ab = wmma_16x16x4_f32(A, Bv, c);
#pragma unroll
    for (int r = 0; r < 8; ++r) {
      float d2 = fmaxf(a2v[r] + b2 - 2.f * ab[r], 0.f);
      if (WITH_IDX) {
        if (d2 < dmin[r]) { dmin[r] = d2; jmin[r] = col; }
      } else {
        dmin[r] = fminf(dmin[r], d2);
      }
    }
  }
}

// argmin across the 16 lanes of each half (same M per 16-lane group)
template <bool WITH_IDX>
DEV_INLINE void wave_reduce_min(float dmin[8], int jmin[8]) {
#pragma unroll
  for (int r = 0; r < 8; ++r) {
#pragma unroll
    for (int off = 1; off < 16; off <<= 1) {
      float od = __shfl_xor(dmin[r], off, 32);
      if (WITH_IDX) {
        int oi = __shfl_xor(jmin[r], off, 32);
        if (od < dmin[r] || (od == dmin[r] && oi < jmin[r])) {
          dmin[r] = od;
          jmin[r] = oi;
        }
      } else {
        dmin[r] = fminf(dmin[r], od);
      }
    }
  }
}

// ---------------------------------------------------------------------------
// Fused 1-NN: one wave owns a 16-row obj tile (A loaded once), scans
// gt (min), recon (argmin) and prior (min) target sets.
// ---------------------------------------------------------------------------
__global__ void __launch_bounds__(256)
knn_fused_kernel(const float* __restrict__ obj, const float* __restrict__ gt,
                 const float* __restrict__ recon,
                 const float* __restrict__ prior, float* __restrict__ dgt,
                 float* __restrict__ drec, int* __restrict__ idxr,
                 float* __restrict__ dpri, int B, int Ns, int V, int NP) {
  const int lane = threadIdx.x & 31;
  const int wave = threadIdx.x >> 5;
  const int gwave = blockIdx.x * (blockDim.x >> 5) + wave;
  const int tilesPerBatch = Ns >> 4;
  const int b = gwave / tilesPerBatch;
  if (b >= B) return;  // wave-uniform
  const int rbase = (gwave % tilesPerBatch) << 4;
  const int l = lane & 15;
  const int half = lane >> 4;

  const float* sb = obj + (size_t)b * Ns * 3;

  // A tile: 16x4 fp32 per ISA layout (lanes 0-15: K=0,1 ; lanes 16-31: K=2,3)
  const float* sp = sb + (size_t)(rbase + l) * 3;
  v2f A;
  A.x = half ? sp[2] : sp[0];
  A.y = half ? 0.f : sp[1];

  // |a|^2 for the 8 rows this lane's D registers cover (M = 8*half + r)
  float a2v[8];
#pragma unroll
  for (int r = 0; r < 8; ++r) {
    const float* q = sb + (size_t)(rbase + half * 8 + r) * 3;
    a2v[r] = q[0] * q[0] + q[1] * q[1] + q[2] * q[2];
  }

  float dmin[8];
  int jmin[8];

  // ---- obj vs gt (min only) ----
  knn_scan<false>(A, a2v, gt + (size_t)b * V * 3, V, l, half, dmin, jmin);
  wave_reduce_min<false>(dmin, jmin);
  if (l == 0) {
#pragma unroll
    for (int r = 0; r < 8; ++r)
      dgt[(size_t)b * Ns + rbase + half * 8 + r] = dmin[r];
  }

  // ---- obj vs recon (argmin) ----
  knn_scan<true>(A, a2v, recon + (size_t)b * V * 3, V, l, half, dmin, jmin);
  wave_reduce_min<true>(dmin, jmin);
  if (l == 0) {
#pragma unroll
    for (int r = 0; r < 8; ++r) {
      int row = rbase + half * 8 + r;
      drec[(size_t)b * Ns + row] = dmin[r];
      idxr[(size_t)b * Ns + row] = jmin[r];
    }
  }

  // ---- obj vs prior (min only) ----
  knn_scan<false>(A, a2v, prior + (size_t)b * NP * 3, NP, l, half, dmin, jmin);
  wave_reduce_min<false>(dmin, jmin);
  if (l == 0) {
#pragma unroll
    for (int r = 0; r < 8; ++r)
      dpri[(size_t)b * Ns + rbase + half * 8 + r] = dmin[r];
  }
}

// ---------------------------------------------------------------------------
// Bidirectional chamfer tiles: recon(V) x gt(V), V not multiple of 16.
// Row minima written directly; column minima via atomicMin on float bits.
// ---------------------------------------------------------------------------
__global__ void __launch_bounds__(256)
cd_kernel(const float* __restrict__ recon, const float* __restrict__ gt,
          float* __restrict__ rowmin, unsigned* __restrict__ colmin, int B,
          int V) {
  const int lane = threadIdx.x & 31;
  const int wave = threadIdx.x >> 5;
  const int gwave = blockIdx.x * (blockDim.x >> 5) + wave;
  const int tilesPerBatch = (V + 15) >> 4;
  const int b = gwave / tilesPerBatch;
  if (b >= B) return;  // wave-uniform
  const int rbase = (gwave % tilesPerBatch) << 4;
  const int l = lane & 15;
  const int half = lane >> 4;

  const float* sb = recon + (size_t)b * V * 3;
  const float* tb = gt + (size_t)b * V * 3;

  const int row = rbase + l;
  const bool rvalid = row < V;
  const float* sp = sb + (size_t)(rvalid ? row : (V - 1)) * 3;
  float sx = rvalid ? sp[0] : 0.f;
  float sy = rvalid ? sp[1] : 0.f;
  float sz = rvalid ? sp[2] : 0.f;
  v2f A;
  A.x = half ? sz : sx;
  A.y = half ? 0.f : sy;

  float a2v[8];
#pragma unroll
  for (int r = 0; r < 8; ++r) {
    int r2 = rbase + half * 8 + r;
    bool v = r2 < V;
    const float* q = sb + (size_t)(v ? r2 : (V - 1)) * 3;
    float d = q[0] * q[0] + q[1] * q[1] + q[2] * q[2];
    a2v[r] = v ? d : __builtin_inff();
  }

  float dmin[8];
#pragma unroll
  for (int r = 0; r < 8; ++r) dmin[r] = __builtin_inff();

  const int full = V >> 4;
  for (int ct = 0; ct <= full; ++ct) {  // last iteration is the tail tile
    const int col = (ct << 4) + l;
    const bool valid = col < V;
    const float* q = tb + (size_t)(valid ? col : (V - 1)) * 3;
    float tx = q[0], ty = q[1], tz = q[2];
    float b2 = valid ? (tx * tx + ty * ty + tz * tz) : __builtin_inff();
    v2f Bv;
    Bv.x = half ? tz : tx;
    Bv.y = half ? 0.f : ty;
    v8f c = {0.f, 0.f, 0.f, 0.f, 0.f, 0.f, 0.f, 0.f};
    v8f ab = wmma_16x16x4_f32(A, Bv, c);

    float colpart = __builtin_inff();
#pragma unroll
    for (int r = 0; r < 8; ++r) {
      float d2 = fmaxf(a2v[r] + b2 - 2.f * ab[r], 0.f);
      dmin[r] = fminf(dmin[r], d2);
      colpart = fminf(colpart, d2);
    }
    // combine the two halves (M 0-7 with M 8-15) for this tile's columns
    colpart = fminf(colpart, __shfl_xor(colpart, 16, 32));
    if (half == 0 && valid)
      atomicMin(&colmin[(size_t)b * V + col], __float_as_uint(colpart));
  }

  // row minima across the 16 lanes of each half
#pragma unroll
  for (int r = 0; r < 8; ++r) {
#pragma unroll
    for (int off = 1; off < 16; off <<= 1)
      dmin[r] = fminf(dmin[r], __shfl_xor(dmin[r], off, 32));
  }
  if (l == 0) {
#pragma unroll
    for (int r = 0; r < 8; ++r) {
      int rw = rbase + half * 8 + r;
      if (rw < V) rowmin[(size_t)b * V + rw] = dmin[r];
    }
  }
}

// ---------------------------------------------------------------------------
// Vertex normals: scatter-add face normals, then normalize.
// ---------------------------------------------------------------------------
__global__ void face_normals(const float* __restrict__ recon,
                             const int* __restrict__ faces,
                             float* __restrict__ vn, int B, int V, int F) {
  int i = blockIdx.x * blockDim.x + threadIdx.x;
  if (i >= B * F) return;
  int b = i / F, f = i % F;
  int i0 = faces[f * 3 + 0], i1 = faces[f * 3 + 1], i2 = faces[f * 3 + 2];
  const float* base = recon + (size_t)b * V * 3;
  float ax = base[i0 * 3 + 0], ay = base[i0 * 3 + 1], az = base[i0 * 3 + 2];
  float bx = base[i1 * 3 + 0], by = base[i1 * 3 + 1], bz = base[i1 * 3 + 2];
  float cx = base[i2 * 3 + 0], cy = base[i2 * 3 + 1], cz = base[i2 * 3 + 2];
  float e1x = bx - ax, e1y = by - ay, e1z = bz - az;
  float e2x = cx - ax, e2y = cy - ay, e2z = cz - az;
  float fx = e1y * e2z - e1z * e2y;
  float fy = e1z * e2x - e1x * e2z;
  float fz = e1x * e2y - e1y * e2x;
  float* vb = vn + (size_t)b * V * 3;
  atomicAdd(&vb[i0 * 3 + 0], fx); atomicAdd(&vb[i0 * 3 + 1], fy); atomicAdd(&vb[i0 * 3 + 2], fz);
  atomicAdd(&vb[i1 * 3 + 0], fx); atomicAdd(&vb[i1 * 3 + 1], fy); atomicAdd(&vb[i1 * 3 + 2], fz);
  atomicAdd(&vb[i2 * 3 + 0], fx); atomicAdd(&vb[i2 * 3 + 1], fy); atomicAdd(&vb[i2 * 3 + 2], fz);
}

__global__ void normalize_vn(float* __restrict__ vn, int n) {
  int i = blockIdx.x * blockDim.x + threadIdx.x;
  if (i >= n) return;
  float x = vn[i * 3 + 0], y = vn[i * 3 + 1], z = vn[i * 3 + 2];
  float inv = 1.f / (sqrtf(x * x + y * y + z * z) + 1e-12f);
  vn[i * 3 + 0] = x * inv; vn[i * 3 + 1] = y * inv; vn[i * 3 + 2] = z * inv;
}

// ---------------------------------------------------------------------------
// Per-point reductions: cmap, consistency, penetration
// acc[0]=sum(cmap? d_prior), acc[1]=n_points, acc[2]=recon&gt count,
// acc[3]=gt count, acc[4]=penetration sum
// ---------------------------------------------------------------------------
__global__ void big_reduce(const float* __restrict__ obj,
                           const float* __restrict__ recon,
                           const float* __restrict__ vn,
                           const float* __restrict__ dgt,
                           const float* __restrict__ drec,
                           const int* __restrict__ idxr,
                           const float* __restrict__ dpri,
                           float* __restrict__ acc, int B, int N1, int V) {
  int i = blockIdx.x * blockDim.x + threadIdx.x;
  float v[5] = {0.f, 0.f, 0.f, 0.f, 0.f};
  if (i < B * N1) {
    int b = i / N1;
    float dr = drec[i], dg = dgt[i];
    bool cmap = dr < 1e-4f;  // 0.01^2
    v[0] = cmap ? dpri[i] : 0.f;
    v[1] = cmap ? 1.f : 0.f;
    bool rc = dr < 2.5e-5f;  // sqrt(dr) < 0.005
    bool gc = dg < 2.5e-5f;
    v[2] = (rc && gc) ? 1.f : 0.f;
    v[3] = gc ? 1.f : 0.f;
    int idx = idxr[i];
    const float* rp = recon + ((size_t)b * V + idx) * 3;
    const float* np = vn + ((size_t)b * V + idx) * 3;
    const float* op = obj + (size_t)i * 3;
    float dot = (rp[0] - op[0]) * np[0] + (rp[1] - op[1]) * np[1] +
                (rp[2] - op[2]) * np[2];
    v[4] = (dot > 0.f) ? dr : 0.f;
  }
  __shared__ float s[256];
#pragma unroll
  for (int q = 0; q < 5; ++q) {
    s[threadIdx.x] = v[q];
    __syncthreads();
    for (int st = 128; st > 0; st >>= 1) {
      if (threadIdx.x < st) s[threadIdx.x] += s[threadIdx.x + st];
      __syncthreads();
    }
    if (threadIdx.x == 0) atomicAdd(&acc[q], s[0]);
    __syncthreads();
  }
}

// ---------------------------------------------------------------------------
// Small reductions: acc[5]=chamfer row+col sums, acc[6]=param sq sum,
// acc[7]=KLD inner sum
// ---------------------------------------------------------------------------
__global__ void misc_reduce(const float* __restrict__ rowmin,
                            const float* __restrict__ colminf,
                            const float* __restrict__ rp,
                            const float* __restrict__ xp,
                            const float* __restrict__ mean,
                            const float* __restrict__ logv,
                            float* __restrict__ acc, int BV, int BP, int BZ) {
  int i = blockIdx.x * blockDim.x + threadIdx.x;
  float v0 = 0.f, v1 = 0.f, v2 = 0.f;
  if (i < BV) v0 = rowmin[i] + colminf[i];
  if (i < BP) { float d = rp[i] - xp[i]; v1 = d * d; }
  if (i < BZ) { float m = mean[i], lv = logv[i]; v2 = 1.f + lv - m * m - expf(lv); }
  __shared__ float s[256];
  float vv[3] = {v0, v1, v2};
#pragma unroll
  for (int q = 0; q < 3; ++q) {
    s[threadIdx.x] = vv[q];
    __syncthreads();
    for (int st = 128; st > 0; st >>= 1) {
      if (threadIdx.x < st) s[threadIdx.x] += s[threadIdx.x + st];
      __syncthreads();
    }
    if (threadIdx.x == 0) atomicAdd(&acc[5 + q], s[0]);
    __syncthreads();
  }
}

__global__ void finalize(const float* __restrict__ acc, float* __restrict__ out,
                         float fB) {
  float recon_loss = acc[5] / fB;  // row+col chamfer sums
  float param_loss = acc[6] / fB;
  float KLD = -0.5f * acc[7] / fB * 10.f;
  float cvae = recon_loss + KLD;
  float cmap_loss = 3000.f * acc[0] / (fB * acc[1]);
  float consistency = -5.f * acc[2] / (acc[3] + 1e-4f);
  float penetr = 100.f * acc[4] / fB;
  out[0] = cvae + 0.1f * param_loss + 1000.f * cmap_loss +
           10.f * consistency + 10.f * penetr;
}

// ---------------------------------------------------------------------------
extern "C" void kernel_launch(void* const* d_in, const int* in_sizes, int n_in,
                              void* d_out, int out_size, void* d_ws,
                              size_t ws_size, hipStream_t stream) {
  (void)in_sizes; (void)n_in; (void)out_size; (void)ws_size;
  const int B = 64, N1 = 8192, V = 778, F = 1538, Z = 64, P = 61, NP = 204;

  const float* obj = (const float*)d_in[0];
  const float* recon = (const float*)d_in[1];
  const float* gt = (const float*)d_in[2];
  const float* meanp = (const float*)d_in[3];
  const float* logv = (const float*)d_in[4];
  const float* rparam = (const float*)d_in[5];
  const float* xparam = (const float*)d_in[6];
  const int* faces = (const int*)d_in[7];

  float* ws = (float*)d_ws;
  size_t off = 0;
  float* dgt = ws + off;    off += (size_t)B * N1;
  float* drec = ws + off;   off += (size_t)B * N1;
  int* idxr = (int*)(ws + off); off += (size_t)B * N1;
  float* dpri = ws + off;   off += (size_t)B * N1;
  float* prior = ws + off;  off += (size_t)B * NP * 3;
  float* vn = ws + off;     off += (size_t)B * V * 3;
  float* rowmin = ws + off; off += (size_t)B * V;
  unsigned* colmin = (unsigned*)(ws + off); off += (size_t)B * V;
  float* acc = ws + off;    off += 16;

  int initN = B * V * 3;
  init_kernel<<<(initN + 255) / 256, 256, 0, stream>>>(vn, colmin, acc, initN,
                                                       B * V);
  gather_prior<<<(B * NP + 255) / 256, 256, 0, stream>>>(recon, prior, B, V, NP);

  int knnBlocks = (B * (N1 / 16)) / 8;  // 8 waves per 256-thread block
  knn_fused_kernel<<<knnBlocks, 256, 0, stream>>>(obj, gt, recon, prior, dgt,
                                                  drec, idxr, dpri, B, N1, V,
                                                  NP);

  int cdWaves = B * ((V + 15) / 16);
  cd_kernel<<<(cdWaves + 7) / 8, 256, 0, stream>>>(recon, gt, rowmin, colmin, B, V);

  face_normals<<<(B * F + 255) / 256, 256, 0, stream>>>(recon, faces, vn, B, V, F);
  normalize_vn<<<(B * V + 255) / 256, 256, 0, stream>>>(vn, B * V);

  big_reduce<<<(B * N1 + 255) / 256, 256, 0, stream>>>(obj, recon, vn, dgt, drec,
                                                       idxr, dpri, acc, B, N1, V);
  misc_reduce<<<(B * V + 255) / 256, 256, 0, stream>>>(
      rowmin, (const float*)colmin, rparam, xparam, meanp, logv, acc, B * V,
      B * P, B * Z);
  finalize<<<1, 1, 0, stream>>>(acc, (float*)d_out, (float)B);
}